// CPAMDec_84172769067181
// MI455X (gfx1250) — compile-verified
//
#include <hip/hip_runtime.h>
#include <math.h>

typedef float v2f __attribute__((ext_vector_type(2)));
typedef float v8f __attribute__((ext_vector_type(8)));

#define N_   8
#define C_   512
#define HW_  4096
#define K_   32
#define C4_  128
#define TP_  64     // pixels per workgroup tile
#define KC_  16     // channel chunk staged in LDS (double-buffered)
#define NCHUNK_ (C_ / KC_)

// ---- CDNA5 async global->LDS copy (ASYNCcnt-tracked, bypasses VGPRs) ------
__device__ __forceinline__ unsigned lds_off(const void* p) {
    // flat LDS pointer: addr[31:0] == LDS byte offset (ISA 10.2 aperture rule)
    return (unsigned)(uintptr_t)p;
}
__device__ __forceinline__ void async_copy_b128(unsigned lds_byte, const float* gaddr) {
    asm volatile("global_load_async_to_lds_b128 %0, %1, off"
                 :: "v"(lds_byte), "v"(gaddr) : "memory");
}
__device__ __forceinline__ void wait_async0() {
    asm volatile("s_wait_asynccnt 0" ::: "memory");
}

// ---------------------------------------------------------------------------
// Kernel 1: k = y @ wk^T + bk  (N,K,C4) ; v = y @ wv^T + bv  (N,K,C)
// One block per (n, kk) row. Tiny GEMMs -> simple float4 dot products.
// ---------------------------------------------------------------------------
__global__ void kv_kernel(const float* __restrict__ y,
                          const float* __restrict__ wk, const float* __restrict__ bk,
                          const float* __restrict__ wv, const float* __restrict__ bv,
                          float* __restrict__ kbuf, float* __restrict__ vbuf) {
    const int n  = blockIdx.x >> 5;   // 0..7
    const int kk = blockIdx.x & 31;   // 0..31
    const int tid = threadIdx.x;

    __shared__ __align__(16) float yrow[C_];
    for (int i = tid; i < C_ / 4; i += 256)
        *(float4*)&yrow[i * 4] = *(const float4*)&y[((size_t)n * K_ + kk) * C_ + i * 4];
    __syncthreads();

    if (tid < C4_) {
        const float* wr = wk + (size_t)tid * C_;
        float acc = 0.f;
        for (int c = 0; c < C_; c += 4) {
            float4 w4 = *(const float4*)(wr + c);
            float4 y4 = *(const float4*)(yrow + c);
            acc += w4.x * y4.x + w4.y * y4.y + w4.z * y4.z + w4.w * y4.w;
        }
        kbuf[((size_t)n * K_ + kk) * C4_ + tid] = acc + bk[tid];
    }

    #pragma unroll
    for (int rep = 0; rep < 2; ++rep) {
        const int cidx = tid + rep * 256;
        const float* wr = wv + (size_t)cidx * C_;
        float acc = 0.f;
        for (int c = 0; c < C_; c += 4) {
            float4 w4 = *(const float4*)(wr + c);
            float4 y4 = *(const float4*)(yrow + c);
            acc += w4.x * y4.x + w4.y * y4.y + w4.z * y4.z + w4.w * y4.w;
        }
        vbuf[((size_t)n * K_ + kk) * C_ + cidx] = acc + bv[cidx];
    }
}

// ---------------------------------------------------------------------------
// Kernel 2: fused q-conv -> energy -> softmax -> out-GEMM -> residual.
// Grid (HW/TP, N), 256 threads = 8 wave32s.
// LDS (floats), unioned across phases (~59.9 KB total):
//   [0     .. 8448)  Qs 64x132 | phase1 double buffers:
//                      Xs0 16x68 @0, Xs1 @1088, Ws0 128x20 @2176, Ws1 @4736
//   [8448  ..12672)  KVs 32x132  (Ks in phase2, Vs chunk in phase4)
//   [12672 ..14976)  Es 64x36    (energy / attn)
// ---------------------------------------------------------------------------
__global__ void __launch_bounds__(256)
attn_kernel(const float* __restrict__ x, const float* __restrict__ wq,
            const float* __restrict__ bq,
            const float* __restrict__ kbuf, const float* __restrict__ vbuf,
            const float* __restrict__ scale, float* __restrict__ out) {
    __shared__ __align__(16) float smem[14976];
    float* Qs  = smem;            // 64 rows (pixel) x 132 (c4, padded)
    float* KVs = smem + 8448;     // 32 rows (k)     x 132
    float* Es  = smem + 12672;    // 64 rows (pixel) x 36

    const int tid  = threadIdx.x;
    const int wave = tid >> 5;
    const int lane = tid & 31;
    const int half = lane >> 4;       // 0: lanes 0-15, 1: lanes 16-31
    const int m    = lane & 15;
    const int n    = blockIdx.y;
    const int p0   = blockIdx.x * TP_;     // linear pixel offset in HW

    // ---------------- Phase 1: Q tile (TP_ x C4_) -----------------
    // 4 Mtiles (pixel) x 8 Ntiles (c4). Wave w: Mtile = w>>1, Ntiles (w&1)*4 ..+3
    const int mt  = wave >> 1;
    const int ntb = (wave & 1) * 4;
    v8f qacc[4] = {v8f(0.f), v8f(0.f), v8f(0.f), v8f(0.f)};

    // async-issue helper (all 256 threads participate):
    //   Xs chunk: 16 ch x 64 px = 256 float4 (1 per thread)
    //   Ws chunk: 128 c4 x 16 ch = 512 float4 (2 per thread)
    auto issue_chunk = [&](int c0, int buf) {
        {
            const int cc = tid >> 4, p4 = tid & 15;
            async_copy_b128(lds_off(&smem[buf * 1088 + cc * 68 + p4 * 4]),
                            x + ((size_t)n * C_ + c0 + cc) * HW_ + p0 + p4 * 4);
        }
        #pragma unroll
        for (int r = 0; r < 2; ++r) {
            const int i = tid + r * 256;
            const int o = i >> 2, c4i = i & 3;
            async_copy_b128(lds_off(&smem[2176 + buf * 2560 + o * 20 + c4i * 4]),
                            wq + (size_t)o * C_ + c0 + c4i * 4);
        }
    };

    issue_chunk(0, 0);
    for (int ci = 0; ci < NCHUNK_; ++ci) {
        wait_async0();        // this wave's async writes (chunk ci) landed in LDS
        __syncthreads();      // ... and everyone else's too
        if (ci + 1 < NCHUNK_) issue_chunk((ci + 1) * KC_, (ci + 1) & 1);

        const float* Xb = smem + (ci & 1) * 1088;
        const float* Wb = smem + 2176 + (ci & 1) * 2560;
        #pragma unroll
        for (int ks = 0; ks < KC_ / 4; ++ks) {
            const int cc = ks * 4 + 2 * half;   // K component base for this lane-half
            v2f a;
            a.x = Xb[(cc + 0) * 68 + mt * 16 + m];
            a.y = Xb[(cc + 1) * 68 + mt * 16 + m];
            #pragma unroll
            for (int j = 0; j < 4; ++j) {
                const v2f b = *(const v2f*)&Wb[((ntb + j) * 16 + m) * 20 + cc];
                qacc[j] = __builtin_amdgcn_wmma_f32_16x16x4_f32(
                    false, a, false, b, (short)0, qacc[j], false, false);
            }
        }
    }
    __syncthreads();   // all waves done reading Xs/Ws before Qs overwrites them

    // bias + spill Q to LDS: Qs[p][c4]
    #pragma unroll
    for (int j = 0; j < 4; ++j) {
        const int c4 = (ntb + j) * 16 + m;
        const float bqv = bq[c4];
        #pragma unroll
        for (int v = 0; v < 8; ++v) {
            const int p = mt * 16 + v + 8 * half;
            Qs[p * 132 + c4] = qacc[j][v] + bqv;
        }
    }
    // stage k_n: KVs[kk][c4] (b128 synchronous; small)
    for (int i = tid; i < K_ * C4_ / 4; i += 256) {
        const int kk = i >> 5, q = i & 31;
        *(float4*)&KVs[kk * 132 + q * 4] =
            *(const float4*)&kbuf[((size_t)n * K_ + kk) * C4_ + q * 4];
    }
    __syncthreads();

    // ---------------- Phase 2: energy (TP_ x K_) ------------------
    {
        const int mt2 = wave & 3;
        const int nt2 = wave >> 2;
        v8f eacc = v8f(0.f);
        #pragma unroll 4
        for (int ks = 0; ks < C4_ / 4; ++ks) {
            const int cc = ks * 4 + 2 * half;
            const v2f a = *(const v2f*)&Qs[(mt2 * 16 + m) * 132 + cc];
            const v2f b = *(const v2f*)&KVs[(nt2 * 16 + m) * 132 + cc];
            eacc = __builtin_amdgcn_wmma_f32_16x16x4_f32(
                false, a, false, b, (short)0, eacc, false, false);
        }
        #pragma unroll
        for (int v = 0; v < 8; ++v) {
            const int p = mt2 * 16 + v + 8 * half;
            Es[p * 36 + nt2 * 16 + m] = eacc[v];
        }
    }
    __syncthreads();

    // ---------------- Phase 3: softmax over K_ --------------------
    if (tid < TP_) {
        float mx = -INFINITY;
        #pragma unroll
        for (int kk = 0; kk < K_; ++kk) mx = fmaxf(mx, Es[tid * 36 + kk]);
        float s = 0.f;
        #pragma unroll
        for (int kk = 0; kk < K_; ++kk) {
            const float e = __expf(Es[tid * 36 + kk] - mx);
            Es[tid * 36 + kk] = e;
            s += e;
        }
        const float inv = 1.0f / s;
        #pragma unroll
        for (int kk = 0; kk < K_; ++kk) Es[tid * 36 + kk] *= inv;
    }
    __syncthreads();

    // ---------------- Phase 4: out = scale * (v^T @ attn^T) + x ---
    // Per 128-channel block: 8 Mtiles (c) x 4 Ntiles (p); wave w -> Mtile w
    const float scl = scale[0];
    for (int cb = 0; cb < C_ / C4_; ++cb) {
        __syncthreads();
        for (int i = tid; i < K_ * C4_ / 4; i += 256) {
            const int kk = i >> 5, q = i & 31;
            *(float4*)&KVs[kk * 132 + q * 4] =
                *(const float4*)&vbuf[((size_t)n * K_ + kk) * C_ + cb * C4_ + q * 4];
        }
        __syncthreads();
        #pragma unroll
        for (int nt = 0; nt < 4; ++nt) {
            v8f oacc = v8f(0.f);
            #pragma unroll
            for (int ks = 0; ks < K_ / 4; ++ks) {
                const int kb = ks * 4 + 2 * half;
                v2f a;
                a.x = KVs[(kb + 0) * 132 + wave * 16 + m];   // V[kk][c_local]
                a.y = KVs[(kb + 1) * 132 + wave * 16 + m];
                const v2f b = *(const v2f*)&Es[(nt * 16 + m) * 36 + kb]; // attn[p][kk]
                oacc = __builtin_amdgcn_wmma_f32_16x16x4_f32(
                    false, a, false, b, (short)0, oacc, false, false);
            }
            const int pg = p0 + nt * 16 + m;
            #pragma unroll
            for (int v = 0; v < 8; ++v) {
                const int c = cb * C4_ + wave * 16 + v + 8 * half;
                const size_t idx = ((size_t)n * C_ + c) * HW_ + pg;
                out[idx] = scl * oacc[v] + x[idx];
            }
        }
    }
}

// ---------------------------------------------------------------------------
extern "C" void kernel_launch(void* const* d_in, const int* in_sizes, int n_in,
                              void* d_out, int out_size, void* d_ws, size_t ws_size,
                              hipStream_t stream) {
    const float* x     = (const float*)d_in[0];
    const float* y     = (const float*)d_in[1];
    const float* wq    = (const float*)d_in[2];
    const float* bq    = (const float*)d_in[3];
    const float* wk    = (const float*)d_in[4];
    const float* bk    = (const float*)d_in[5];
    const float* wv    = (const float*)d_in[6];
    const float* bv    = (const float*)d_in[7];
    const float* scale = (const float*)d_in[8];
    float* out  = (float*)d_out;
    float* kbuf = (float*)d_ws;                      // N*K*C4 floats
    float* vbuf = kbuf + (size_t)N_ * K_ * C4_;      // N*K*C  floats

    kv_kernel<<<dim3(N_ * K_), dim3(256), 0, stream>>>(y, wk, bk, wv, bv, kbuf, vbuf);
    attn_kernel<<<dim3(HW_ / TP_, N_), dim3(256), 0, stream>>>(x, wq, bq, kbuf, vbuf, scale, out);
}